// XLMRCorefModel_57080115364308
// MI455X (gfx1250) — compile-verified
//
#include <hip/hip_runtime.h>
#include <math.h>

typedef __attribute__((ext_vector_type(2))) float v2f;
typedef __attribute__((ext_vector_type(8))) float v8f;

#define B_  2
#define L_  512
#define H_  768
#define W_  10
#define K_  512
#define NL_ 50
#define E_  2048
#define NSPAN (L_ * W_)        // 5120
#define H3  (3 * H_)           // 2304
#define MROWS (B_ * NSPAN)     // 10240
#define NEGC (-1000000000.0f)

// ---- CDNA5 async global->LDS copy (ASYNCcnt-tracked, bypasses VGPRs) -------
__device__ __forceinline__ void async_load_b128(const float* g, const void* lds) {
  // low 32 bits of a generic pointer to LDS == LDS byte address (flat aperture)
  unsigned l = (unsigned)(unsigned long long)(uintptr_t)lds;
  asm volatile("global_load_async_to_lds_b128 %0, %1, off"
               :: "v"(l), "v"(g) : "memory");
}
__device__ __forceinline__ void wait_asynccnt0() {
  asm volatile("s_wait_asynccnt 0x0" ::: "memory");
}

// ---------------------------------------------------------------------------
// 1) GCN: per-edge  y = gcn_w[lab] @ src_emb + gcn_b[lab], scatter-add to tgt
// ---------------------------------------------------------------------------
__global__ __launch_bounds__(256) void gcn_edge_kernel(
    const float* __restrict__ emb0, const int* __restrict__ edges,
    const float* __restrict__ gcn_w, const float* __restrict__ gcn_b,
    float* __restrict__ agg, float* __restrict__ deg) {
  int be = blockIdx.x;                 // b*E + e
  int b  = be / E_;
  int src = edges[be * 3 + 0];
  int tgt = edges[be * 3 + 1];
  int lab = edges[be * 3 + 2];
  bool emask = (lab != -100) && (src < L_) && (tgt < L_);
  if (!emask) return;                  // zero contribution, zero deg
  int sc = min(max(src, 0), L_ - 1);
  int lc = min(max(lab, 0), NL_ - 1);

  __shared__ float x[H_];
  int tid = threadIdx.x;
  for (int h = tid; h < H_; h += 256)
    x[h] = emb0[((size_t)b * L_ + sc) * H_ + h];
  __syncthreads();

  int wave = tid >> 5, lane = tid & 31;
  const float* Wl = gcn_w + (size_t)lc * H_ * H_;
  for (int r = wave; r < H_; r += 8) {
    const float* wr = Wl + (size_t)r * H_;
    float acc = 0.f;
    for (int k = lane; k < H_; k += 32) acc += wr[k] * x[k];
    for (int off = 16; off; off >>= 1) acc += __shfl_down(acc, off, 32);
    if (lane == 0)
      atomicAdd(&agg[((size_t)b * L_ + tgt) * H_ + r], acc + gcn_b[lc * H_ + r]);
  }
  if (tid == 0) atomicAdd(&deg[b * L_ + tgt], 1.0f);
}

// ---------------------------------------------------------------------------
// 2) emb = embeddings + relu(agg / clip(deg,1))
// ---------------------------------------------------------------------------
__global__ __launch_bounds__(256) void emb_kernel(
    const float* __restrict__ emb0, const float* __restrict__ agg,
    const float* __restrict__ deg, float* __restrict__ emb) {
  int i = blockIdx.x * 256 + threadIdx.x;
  if (i >= B_ * L_ * H_) return;
  float d = fmaxf(deg[i / H_], 1.0f);
  emb[i] = emb0[i] + fmaxf(agg[i] / d, 0.0f);
}

// ---------------------------------------------------------------------------
// 3) attn_logits[b,p] = emb[b,p,:]·attn_w + attn_b   (wave per row)
// ---------------------------------------------------------------------------
__global__ __launch_bounds__(256) void attn_kernel(
    const float* __restrict__ emb, const float* __restrict__ attn_w,
    const float* __restrict__ attn_b, float* __restrict__ logits) {
  int wave = threadIdx.x >> 5, lane = threadIdx.x & 31;
  int row = blockIdx.x * 8 + wave;     // 0 .. B*L-1
  const float* er = emb + (size_t)row * H_;
  float p = 0.f;
  for (int c = lane; c < H_; c += 32) p += er[c] * attn_w[c];
  for (int off = 16; off; off >>= 1) p += __shfl_down(p, off, 32);
  if (lane == 0) logits[row] = p + attn_b[0];
}

// ---------------------------------------------------------------------------
// 4) span_rep[b,n] = [start | end | softmax-pooled]   (block per span)
// ---------------------------------------------------------------------------
__global__ __launch_bounds__(256) void span_kernel(
    const float* __restrict__ emb, const float* __restrict__ logits,
    float* __restrict__ span_rep) {
  int bn = blockIdx.x;                 // b*NSPAN + n
  int b = bn / NSPAN, n = bn % NSPAN;
  int s = n / W_, w = n % W_;

  float aw[W_];
  float mx = NEGC;
  #pragma unroll
  for (int ww = 0; ww < W_; ++ww) {
    int pos = s + ww;
    bool m = (ww <= w) && (pos < L_);
    float v = m ? logits[b * L_ + min(pos, L_ - 1)] : NEGC;
    aw[ww] = v;
    mx = fmaxf(mx, v);
  }
  float sum = 0.f;
  #pragma unroll
  for (int ww = 0; ww < W_; ++ww) { aw[ww] = expf(aw[ww] - mx); sum += aw[ww]; }
  float inv = 1.0f / sum;
  #pragma unroll
  for (int ww = 0; ww < W_; ++ww) aw[ww] *= inv;

  const float* Eb = emb + (size_t)b * L_ * H_;
  int epos = min(s + w, L_ - 1);
  size_t ro = (size_t)bn * H3;
  for (int h = threadIdx.x; h < H_; h += 256) {
    float st = Eb[s * H_ + h];
    float en = Eb[epos * H_ + h];
    float pl = 0.f;
    #pragma unroll
    for (int ww = 0; ww < W_; ++ww)
      pl += aw[ww] * Eb[min(s + ww, L_ - 1) * H_ + h];
    span_rep[ro + h]          = st;
    span_rep[ro + H_ + h]     = en;
    span_rep[ro + 2 * H_ + h] = pl;
  }
}

// ---------------------------------------------------------------------------
// 5) Mention GEMM (WMMA f32 16x16x4), double-buffered async global->LDS,
//    fused bias+ReLU+(·m_w2) reduction.
//    C[m,o] = sum_k span_rep[m,k] * m_w1[o,k];  m_scores[m] += relu(C+b1)·w2
// ---------------------------------------------------------------------------
#define BM 128
#define BN 64
#define KC 32
#define KPAD (KC + 4)                  // stride 36: 16B aligned, conflict-free
#define NCHUNK (H3 / KC)               // 72

__global__ __launch_bounds__(256) void mention_gemm_kernel(
    const float* __restrict__ A, const float* __restrict__ m_w1,
    const float* __restrict__ m_b1, const float* __restrict__ m_w2,
    float* __restrict__ m_scores) {
  __shared__ __align__(16) float sA[2][BM][KPAD];  // 36.9 KB
  __shared__ __align__(16) float sB[2][BN][KPAD];  // 18.4 KB  sB[n][k]=m_w1[n][k]

  int tid = threadIdx.x;
  int lane = tid & 31, wave = tid >> 5;
  int wr = wave >> 1, wc = wave & 1;   // wave grid 4x2 of 32x32 tiles
  int blockM = blockIdx.x * BM;        // 80 tiles (10240/128 exact)
  int blockN = blockIdx.y * BN;        // 36 tiles (2304/64 exact)

  // per-thread staging coordinates (A: 4 x b128, B: 2 x b128 per chunk)
  int aRow[4], aC4[4];
  #pragma unroll
  for (int i = 0; i < 4; ++i) {
    int idx = tid + i * 256;           // 0..1023
    aRow[i] = idx >> 3;
    aC4[i]  = (idx & 7) << 2;
  }
  int bRow[2], bC4[2];
  #pragma unroll
  for (int i = 0; i < 2; ++i) {
    int idx = tid + i * 256;           // 0..511
    bRow[i] = idx >> 3;
    bC4[i]  = (idx & 7) << 2;
  }

  v8f acc00 = {}, acc01 = {}, acc10 = {}, acc11 = {};

  int aRow0 = wr * 32 + (lane & 15);
  int aRow1 = aRow0 + 16;
  int kHalf = (lane >> 4) << 1;        // 0 for lanes 0-15, 2 for 16-31
  int bCol0 = wc * 32 + (lane & 15);
  int bCol1 = bCol0 + 16;

  // prefetch chunk 0 into buffer 0
  #pragma unroll
  for (int i = 0; i < 4; ++i)
    async_load_b128(A + (size_t)(blockM + aRow[i]) * H3 + aC4[i],
                    &sA[0][aRow[i]][aC4[i]]);
  #pragma unroll
  for (int i = 0; i < 2; ++i)
    async_load_b128(m_w1 + (size_t)(blockN + bRow[i]) * H3 + bC4[i],
                    &sB[0][bRow[i]][bC4[i]]);
  wait_asynccnt0();
  __syncthreads();

  for (int c = 0; c < NCHUNK; ++c) {
    int cur = c & 1;
    if (c + 1 < NCHUNK) {              // issue next chunk into the other buffer
      int kt = (c + 1) * KC;
      #pragma unroll
      for (int i = 0; i < 4; ++i)
        async_load_b128(A + (size_t)(blockM + aRow[i]) * H3 + kt + aC4[i],
                        &sA[cur ^ 1][aRow[i]][aC4[i]]);
      #pragma unroll
      for (int i = 0; i < 2; ++i)
        async_load_b128(m_w1 + (size_t)(blockN + bRow[i]) * H3 + kt + bC4[i],
                        &sB[cur ^ 1][bRow[i]][bC4[i]]);
    }

    #pragma unroll
    for (int kk = 0; kk < KC; kk += 4) {
      int ka = kk + kHalf;
      v2f a0 = *(const v2f*)&sA[cur][aRow0][ka];   // A[m,k..k+1]
      v2f a1 = *(const v2f*)&sA[cur][aRow1][ka];
      v2f b0 = *(const v2f*)&sB[cur][bCol0][ka];   // B[k..k+1,n]
      v2f b1 = *(const v2f*)&sB[cur][bCol1][ka];
      acc00 = __builtin_amdgcn_wmma_f32_16x16x4_f32(false, a0, false, b0, (short)0, acc00, false, false);
      acc01 = __builtin_amdgcn_wmma_f32_16x16x4_f32(false, a0, false, b1, (short)0, acc01, false, false);
      acc10 = __builtin_amdgcn_wmma_f32_16x16x4_f32(false, a1, false, b0, (short)0, acc10, false, false);
      acc11 = __builtin_amdgcn_wmma_f32_16x16x4_f32(false, a1, false, b1, (short)0, acc11, false, false);
    }

    wait_asynccnt0();                  // own async loads for next buffer done
    __syncthreads();                   // publish buffer + protect WAR on cur^1
  }

  // Epilogue: v = relu(C + b1) * w2, reduce over the 16 column lanes, atomicAdd
  int col = lane & 15, half = lane >> 4;
  int c0 = blockN + wc * 32 + col;
  int c1 = c0 + 16;
  float b1v0 = m_b1[c0], b1v1 = m_b1[c1];
  float w2v0 = m_w2[c0], w2v1 = m_w2[c1];
  #pragma unroll
  for (int i = 0; i < 2; ++i) {
    v8f accL = i ? acc10 : acc00;
    v8f accR = i ? acc11 : acc01;
    int rowBase = blockM + wr * 32 + i * 16 + half * 8;
    #pragma unroll
    for (int r = 0; r < 8; ++r) {
      float v = fmaxf(accL[r] + b1v0, 0.f) * w2v0 +
                fmaxf(accR[r] + b1v1, 0.f) * w2v1;
      v += __shfl_xor(v, 1, 32);
      v += __shfl_xor(v, 2, 32);
      v += __shfl_xor(v, 4, 32);
      v += __shfl_xor(v, 8, 32);
      if (col == 0) atomicAdd(&m_scores[rowBase + r], v);
    }
  }
}

// ---------------------------------------------------------------------------
// 6) Stable descending top-K by iterative argmax (matches jax.lax.top_k order)
// ---------------------------------------------------------------------------
__global__ __launch_bounds__(256) void topk_kernel(
    const float* __restrict__ m_scores, int* __restrict__ topk_idx) {
  __shared__ float sc[NSPAN];
  __shared__ float redV[256];
  __shared__ int   redI[256];
  int b = blockIdx.x, tid = threadIdx.x;
  for (int n = tid; n < NSPAN; n += 256) {
    int s = n / W_, w = n % W_;
    sc[n] = (s + w < L_) ? m_scores[b * NSPAN + n] : NEGC;
  }
  __syncthreads();
  for (int k = 0; k < K_; ++k) {
    float best = -INFINITY; int bi = 0x7fffffff;
    for (int n = tid; n < NSPAN; n += 256) {
      float v = sc[n];
      if (v > best || (v == best && n < bi)) { best = v; bi = n; }
    }
    redV[tid] = best; redI[tid] = bi;
    __syncthreads();
    for (int off = 128; off; off >>= 1) {
      if (tid < off) {
        float v2 = redV[tid + off]; int i2 = redI[tid + off];
        if (v2 > redV[tid] || (v2 == redV[tid] && i2 < redI[tid])) {
          redV[tid] = v2; redI[tid] = i2;
        }
      }
      __syncthreads();
    }
    if (tid == 0) { topk_idx[b * K_ + k] = redI[0]; sc[redI[0]] = -INFINITY; }
    __syncthreads();
  }
}

// ---------------------------------------------------------------------------
// 7) sa/sb: dots of pruned span_rep rows with ant_w halves
// ---------------------------------------------------------------------------
__global__ __launch_bounds__(256) void ant_prep_kernel(
    const float* __restrict__ span_rep, const int* __restrict__ topk_idx,
    const float* __restrict__ ant_w, float* __restrict__ sa,
    float* __restrict__ sb) {
  int bi = blockIdx.x;                 // b*K + i
  int b = bi / K_;
  int n = topk_idx[bi];
  const float* row = span_rep + ((size_t)b * NSPAN + n) * H3;
  int tid = threadIdx.x, lane = tid & 31, wave = tid >> 5;
  float pa = 0.f, pb = 0.f;
  for (int c = tid; c < H3; c += 256) {
    float x = row[c];
    pa += x * ant_w[c];
    pb += x * ant_w[H3 + c];
  }
  for (int off = 16; off; off >>= 1) {
    pa += __shfl_down(pa, off, 32);
    pb += __shfl_down(pb, off, 32);
  }
  __shared__ float ra[8], rb[8];
  if (lane == 0) { ra[wave] = pa; rb[wave] = pb; }
  __syncthreads();
  if (tid == 0) {
    float s1 = 0.f, s2 = 0.f;
    for (int w = 0; w < 8; ++w) { s1 += ra[w]; s2 += rb[w]; }
    sa[bi] = s1; sb[bi] = s2;
  }
}

// ---------------------------------------------------------------------------
// 8) ant_scores[b,i,j] = sb[b,i] + sa[b,j] + ant_b
// ---------------------------------------------------------------------------
__global__ __launch_bounds__(256) void ant_out_kernel(
    const float* __restrict__ sa, const float* __restrict__ sb,
    const float* __restrict__ ant_b, float* __restrict__ out) {
  int idx = blockIdx.x * 256 + threadIdx.x;
  if (idx >= B_ * K_ * K_) return;
  int b = idx / (K_ * K_);
  int r = idx % (K_ * K_);
  int i = r / K_, j = r % K_;
  out[idx] = sb[b * K_ + i] + sa[b * K_ + j] + ant_b[0];
}

// ---------------------------------------------------------------------------
extern "C" void kernel_launch(void* const* d_in, const int* in_sizes, int n_in,
                              void* d_out, int out_size, void* d_ws, size_t ws_size,
                              hipStream_t stream) {
  const float* embeddings = (const float*)d_in[0];
  const int*   edges      = (const int*)d_in[1];
  const float* gcn_w      = (const float*)d_in[2];
  const float* gcn_b      = (const float*)d_in[3];
  const float* attn_w     = (const float*)d_in[4];
  const float* attn_b     = (const float*)d_in[5];
  const float* m_w1       = (const float*)d_in[6];
  const float* m_b1       = (const float*)d_in[7];
  const float* m_w2       = (const float*)d_in[8];
  // d_in[9] = m_b2: constant shift of valid scores, does not change top-k order
  const float* ant_w      = (const float*)d_in[10];
  const float* ant_b      = (const float*)d_in[11];
  float* out = (float*)d_out;

  float* ws = (float*)d_ws;
  const size_t nEmb = (size_t)B_ * L_ * H_;        // 786432
  float* agg      = ws;
  float* deg      = agg + nEmb;                    // 1024
  float* m_scores = deg + (B_ * L_);               // 10240
  float* emb      = m_scores + MROWS;              // 786432
  float* logits   = emb + nEmb;                    // 1024
  float* span_rep = logits + (B_ * L_);            // 23,592,960
  float* sa       = span_rep + (size_t)MROWS * H3; // 1024
  float* sb       = sa + (B_ * K_);                // 1024
  int*   topk     = (int*)(sb + (B_ * K_));        // 1024 ints

  // zero accumulation regions (agg|deg|m_scores are contiguous)
  hipMemsetAsync(agg, 0, (nEmb + B_ * L_ + MROWS) * sizeof(float), stream);

  gcn_edge_kernel<<<B_ * E_, 256, 0, stream>>>(embeddings, edges, gcn_w, gcn_b,
                                               agg, deg);
  emb_kernel<<<(int)(nEmb / 256), 256, 0, stream>>>(embeddings, agg, deg, emb);
  attn_kernel<<<(B_ * L_) / 8, 256, 0, stream>>>(emb, attn_w, attn_b, logits);
  span_kernel<<<MROWS, 256, 0, stream>>>(emb, logits, span_rep);
  {
    dim3 grid(MROWS / BM, H3 / BN);   // 80 x 36
    mention_gemm_kernel<<<grid, 256, 0, stream>>>(span_rep, m_w1, m_b1, m_w2,
                                                  m_scores);
  }
  topk_kernel<<<B_, 256, 0, stream>>>(m_scores, topk);
  ant_prep_kernel<<<B_ * K_, 256, 0, stream>>>(span_rep, topk, ant_w, sa, sb);
  ant_out_kernel<<<(B_ * K_ * K_) / 256, 256, 0, stream>>>(sa, sb, ant_b, out);
}